// CapsuleLayer_53626961658065
// MI455X (gfx1250) — compile-verified
//
#include <hip/hip_runtime.h>

#define NCLS  32
#define NCAP  1152
#define ICH   32
#define OCH   32
#define NBATCH 64

typedef float v2f __attribute__((ext_vector_type(2)));
typedef float v8f __attribute__((ext_vector_type(8)));

// ---------------------------------------------------------------------------
// Phase 1: priors[c][b][n][o] = sum_i x[b][n][i] * W[c][n][i][o]
// One wave per (c,n): D[64x32] = A[64x32] @ B[32x32] via V_WMMA_F32_16X16X4_F32
//   - 4 M-tiles (batch), 2 N-tiles (o), 8 K-steps of 4  => 64 WMMA / wave
// Block = 8 waves handling 8 consecutive n for one class; x tile staged in LDS.
// ---------------------------------------------------------------------------
__global__ __launch_bounds__(256) void caps_priors_wmma(
    const float* __restrict__ x,        // [B, N, I]
    const float* __restrict__ W,        // [C, N, I, O]
    float* __restrict__ priors)         // [C, B, N, O]
{
    __shared__ __align__(16) float xs[8 * NBATCH * ICH];   // 64 KB

    const int tid    = threadIdx.x;
    const int lane   = tid & 31;
    const int w      = tid >> 5;                 // wave id 0..7
    const int c      = blockIdx.x / (NCAP / 8);
    const int n_base = (blockIdx.x % (NCAP / 8)) * 8;

    // Cooperative, coalesced load of x[:, n_base..n_base+7, :] into LDS.
    const float4* x4 = (const float4*)x;
    for (int k = tid; k < 8 * NBATCH * ICH / 4; k += 256) {
        int nl = k >> 9;            // 512 float4 per n
        int r  = k & 511;
        int b  = r >> 3;
        int i4 = r & 7;
        float4 v = x4[(size_t)b * (NCAP * ICH / 4) + (size_t)(n_base + nl) * (ICH / 4) + i4];
        *(float4*)&xs[nl * (NBATCH * ICH) + b * ICH + i4 * 4] = v;
    }
    __syncthreads();

    const int    n   = n_base + w;
    const float* Wn  = W + ((size_t)c * NCAP + n) * (ICH * OCH);
    const float* xw  = &xs[w * (NBATCH * ICH)];
    const int    h   = lane >> 4;    // k-pair half (0/1)
    const int    l16 = lane & 15;

    v8f acc[4][2] = {};              // [m-tile][n-tile], 16x16 f32 C/D frags

    for (int s = 0; s < 8; ++s) {    // K steps of 4
        const int krow = 4 * s + 2 * h;
        // B frags: B[k,o] -> lane l16 = o%16, VGPR j = k offset within pair
        v2f bfr[2];
        for (int nt = 0; nt < 2; ++nt) {
            v2f bv;
            bv.x = Wn[(krow + 0) * OCH + nt * 16 + l16];
            bv.y = Wn[(krow + 1) * OCH + nt * 16 + l16];
            bfr[nt] = bv;
        }
        for (int mt = 0; mt < 4; ++mt) {
            // A frag: A[m,k] -> lane l16 = m, k = krow + j (contiguous pair in LDS)
            v2f av = *(const v2f*)&xw[(mt * 16 + l16) * ICH + krow];
            acc[mt][0] = __builtin_amdgcn_wmma_f32_16x16x4_f32(
                false, av, false, bfr[0], (short)0, acc[mt][0], false, false);
            acc[mt][1] = __builtin_amdgcn_wmma_f32_16x16x4_f32(
                false, av, false, bfr[1], (short)0, acc[mt][1], false, false);
        }
    }

    // D layout: element [v + 8*h][l16] of the 16x16 tile.
    for (int mt = 0; mt < 4; ++mt)
        for (int nt = 0; nt < 2; ++nt)
            for (int v = 0; v < 8; ++v) {
                int b = mt * 16 + v + 8 * h;
                int o = nt * 16 + l16;
                priors[(((size_t)c * NBATCH + b) * NCAP + n) * OCH + o] = acc[mt][nt][v];
            }
}

// ---------------------------------------------------------------------------
// Phase 2: dynamic routing, fully fused per (c,b) in LDS (147 KB priors slice).
// ---------------------------------------------------------------------------
__global__ __launch_bounds__(256) void caps_routing(
    const float* __restrict__ priors,   // [C, B, N, O]
    float* __restrict__ out)            // [C, B, 1, 1, O]
{
    __shared__ __align__(16) float pr[NCAP * OCH];   // 147456 B
    __shared__ float blog[NCAP];                     // routing logits (per n)
    __shared__ float en[NCAP];                       // exp(blog - max)
    __shared__ float red[256];                       // reductions / s partials
    __shared__ __align__(16) float vout[OCH];        // squashed output

    const int tid = threadIdx.x;
    const int c   = blockIdx.x >> 6;
    const int b   = blockIdx.x & 63;

    // Single streaming read of priors slice into LDS.
    const float4* s4 = (const float4*)(priors + ((size_t)c * NBATCH + b) * (NCAP * OCH));
    for (int k = tid; k < NCAP * OCH / 4; k += 256)
        *(float4*)&pr[k * 4] = s4[k];
    for (int nn = tid; nn < NCAP; nn += 256) blog[nn] = 0.0f;
    __syncthreads();

    const int o = tid & 31;
    const int g = tid >> 5;

    for (int it = 0; it < 3; ++it) {
        // --- softmax over n: max ---
        float m = -1e30f;
        for (int nn = tid; nn < NCAP; nn += 256) m = fmaxf(m, blog[nn]);
        red[tid] = m; __syncthreads();
        for (int off = 128; off > 0; off >>= 1) {
            if (tid < off) red[tid] = fmaxf(red[tid], red[tid + off]);
            __syncthreads();
        }
        const float mx = red[0]; __syncthreads();

        // --- exp + partition function ---
        float z = 0.0f;
        for (int nn = tid; nn < NCAP; nn += 256) {
            float e = __expf(blog[nn] - mx);
            en[nn] = e;
            z += e;
        }
        red[tid] = z; __syncthreads();
        for (int off = 128; off > 0; off >>= 1) {
            if (tid < off) red[tid] += red[tid + off];
            __syncthreads();
        }
        const float Z = red[0]; __syncthreads();

        // --- s[o] = (1/Z) * sum_n en[n] * pr[n][o], partials over 8 n-groups ---
        float acc = 0.0f;
        for (int nn = g; nn < NCAP; nn += 8) acc += en[nn] * pr[nn * OCH + o];
        red[g * 32 + o] = acc; __syncthreads();

        if (tid < 32) {                       // wave 0 finishes s + squash
            float s = 0.0f;
            for (int gg = 0; gg < 8; ++gg) s += red[gg * 32 + tid];
            s /= Z;
            float sq = s * s;
            for (int msk = 16; msk > 0; msk >>= 1) sq += __shfl_xor(sq, msk, 32);
            float scale = (sq / (1.0f + sq)) * rsqrtf(sq);
            float v = s * scale;
            vout[tid] = v;
            if (it == 2)
                out[((size_t)c * NBATCH + b) * OCH + tid] = v;
        }
        __syncthreads();

        // --- delta[n] = <pr[n,:], vout>; blog += delta ---
        if (it < 2) {
            const float4* vv = (const float4*)vout;
            for (int nn = tid; nn < NCAP; nn += 256) {
                const float4* row = (const float4*)&pr[nn * OCH];
                float d = 0.0f;
                for (int q = 0; q < 8; ++q) {
                    float4 p = row[q], w4 = vv[q];
                    d += p.x * w4.x + p.y * w4.y + p.z * w4.z + p.w * w4.w;
                }
                blog[nn] += d;
            }
            __syncthreads();
        }
    }
}

extern "C" void kernel_launch(void* const* d_in, const int* in_sizes, int n_in,
                              void* d_out, int out_size, void* d_ws, size_t ws_size,
                              hipStream_t stream) {
    (void)in_sizes; (void)n_in; (void)out_size; (void)ws_size;

    const float* x = (const float*)d_in[0];        // [64, 1152, 32] f32
    const float* W = (const float*)d_in[1];        // [32, 1152, 32, 32] f32
    float* out     = (float*)d_out;                // [32, 64, 1, 1, 32] f32
    float* priors  = (float*)d_ws;                 // needs C*B*N*O*4 = 302 MB scratch

    // Phase 1: 32 classes * 144 n-octets, 8 waves per block (one (c,n) per wave)
    caps_priors_wmma<<<dim3(NCLS * (NCAP / 8)), dim3(256), 0, stream>>>(x, W, priors);
    // Phase 2: one workgroup per (c,b)
    caps_routing<<<dim3(NCLS * NBATCH), dim3(256), 0, stream>>>(priors, out);
}